// CorrTorch_unfold_76115410419878
// MI455X (gfx1250) — compile-verified
//
#include <hip/hip_runtime.h>
#include <stdint.h>

// Problem constants (B=1, C=32, D=32, H=128, W=128 per reference).
#define CD 32
#define DD 32
#define HD 128
#define WD 128
#define YROW 140   // padded LDS row width in floats (16B-multiple, bank skew 12)

// ---- CDNA5 async global->LDS copy (16 bytes per lane), tracked by ASYNCcnt ----
__device__ __forceinline__ void async_ld16(uint32_t lds_byte_addr, const float* g) {
  asm volatile("global_load_async_to_lds_b128 %0, %1, off"
               :
               : "v"(lds_byte_addr), "v"((uint64_t)(uintptr_t)g)
               : "memory");
}
__device__ __forceinline__ void wait_async_le8() {
  asm volatile("s_wait_asynccnt 8" ::: "memory");
}
__device__ __forceinline__ void wait_async_0() {
  asm volatile("s_wait_asynccnt 0" ::: "memory");
}

extern "C" __global__ __launch_bounds__(128)
void corr_unfold_kernel(const float* __restrict__ xg,
                        const float* __restrict__ yg,
                        float* __restrict__ outg)
{
  // Double-buffered y tile: 32 rows x 140 floats per buffer (35,840 bytes total).
  __shared__ float yb[2][32 * YROW];

  const int t  = threadIdx.x;            // 0..127  == w'
  const int hp = blockIdx.x & (HD - 1);  // h'
  const int d  = blockIdx.x >> 7;        // depth slice
  const int hq = hp >> 2;
  const int hm = hp & 3;
  const int dep = (d > 0) ? (d - 1) : 0; // y depth index (d==0 -> all-zero outputs)
  const bool dok = (d > 0);

  // x column for this (d, h', w'=t): 32 channels into registers, coalesced loads.
  float xreg[CD];
#pragma unroll
  for (int c = 0; c < CD; ++c)
    xreg[c] = xg[(((size_t)c * DD + d) * HD + hp) * WD + t];

  const int f4   = (t & 31) * 4;  // starting float column of this lane's 16B chunk
  const int rsub = t >> 5;        // wave id 0..3

  // Stage y window for chunk kk into buffer kk&1.
  auto stage = [&](int kk) {
    const int q  = kk * 32 + hq;
    const int c  = q / 9;
    const int k  = q - 9 * c;
    const int kh = k / 3;
    const int hb = 32 * hm + kh - 1;           // first y row of the window
    float* buf = yb[kk & 1];

    // Zero the pad words of each row: float4s at word offsets 0, 132, 136.
    // (word 3 == y col -1, word 132 == y col 128 -> the zero-padding halo)
    if (t < 96) {
      const int r   = t / 3;
      const int sel = t - 3 * r;
      const int off = (sel == 0) ? 0 : ((sel == 1) ? 132 : 136);
      *(float4*)&buf[r * YROW + off] = make_float4(0.f, 0.f, 0.f, 0.f);
    }

    const float* plane = yg + (((size_t)c * DD + dep) * HD) * WD;
    // 32 rows x 32 float4 = 1024 async 16B copies; 8 per thread, row-uniform per wave.
#pragma unroll
    for (int j = 0; j < 8; ++j) {
      const int r = j * 4 + rsub;
      int hr = hb + r;
      hr = (hr < 0) ? 0 : ((hr > HD - 1) ? (HD - 1) : hr);  // clamp; OOB rows never read
      async_ld16((uint32_t)(uintptr_t)&buf[r * YROW + 4 + f4],
                 plane + (size_t)hr * WD + f4);
    }
  };

  stage(0);

  const int wq = t >> 2;
  const int wm = t & 3;

  for (int kk = 0; kk < 9; ++kk) {
    if (kk < 8) { stage(kk + 1); wait_async_le8(); }  // newest 8 pending = chunk kk+1
    else        { wait_async_0(); }
    __syncthreads();   // all waves' async data + ds edge-zeros for chunk kk visible

    const int q  = kk * 32 + hq;
    const int k  = q % 9;
    const int kh = k / 3;
    const int kw = k - 3 * kh;
    const int hb = 32 * hm + kh - 1;
    const int hr = hb + wq;    // the single y row this output uses

    float o = 0.f;
    if (dok && hr >= 0 && hr < HD) {
      const float* row = &yb[kk & 1][wq * YROW + 3 + 32 * wm + kw];
      float s0 = 0.f, s1 = 0.f, s2 = 0.f, s3 = 0.f;
#pragma unroll
      for (int c = 0; c < CD; c += 4) {
        s0 = fmaf(row[c + 0], xreg[c + 0], s0);
        s1 = fmaf(row[c + 1], xreg[c + 1], s1);
        s2 = fmaf(row[c + 2], xreg[c + 2], s2);
        s3 = fmaf(row[c + 3], xreg[c + 3], s3);
      }
      o = (s0 + s1) + (s2 + s3);
    }
    o = (o > 0.f) ? o : 0.2f * o;   // leaky_relu, slope 0.2 (also maps 0 -> 0 for d==0/OOB)
    outg[(((size_t)kk * DD + d) * HD + hp) * WD + t] = o;

    __syncthreads();   // protect buffer kk&1 before stage(kk+2) overwrites it
  }
}

extern "C" void kernel_launch(void* const* d_in, const int* in_sizes, int n_in,
                              void* d_out, int out_size, void* d_ws, size_t ws_size,
                              hipStream_t stream) {
  (void)in_sizes; (void)n_in; (void)d_ws; (void)ws_size; (void)out_size;
  const float* x = (const float*)d_in[0];
  const float* y = (const float*)d_in[1];
  float* out = (float*)d_out;
  corr_unfold_kernel<<<dim3(DD * HD), dim3(128), 0, stream>>>(x, y, out);
}